// EERNNSeqNet_979252543893
// MI455X (gfx1250) — compile-verified
//
#include <hip/hip_runtime.h>
#include <hip/hip_bf16.h>
#include <math.h>

// Problem constants from the reference
#define QDIM 256      // question size
#define HDIM 64       // hidden size
#define KSEL 64       // top-k

typedef __attribute__((ext_vector_type(2))) float v2f;
typedef __attribute__((ext_vector_type(8))) float v8f;

// ---------------------------------------------------------------------------
// Sortable key: larger float -> larger unsigned key (total order, desc top-k)
// ---------------------------------------------------------------------------
__device__ __forceinline__ unsigned int keyOf(float f) {
    unsigned int u = __float_as_uint(f);
    return (u & 0x80000000u) ? ~u : (u | 0x80000000u);
}

// ---------------------------------------------------------------------------
// init: zero 256-bin histogram + radix-select state
// state[0]=prefix  state[1]=remainingK  state[2]=countGreater
// state[3]=cGreater counter  state[4]=cEqual counter
// ---------------------------------------------------------------------------
__global__ void eernn_init(unsigned int* __restrict__ hist,
                           unsigned int* __restrict__ state) {
    int t = threadIdx.x;
    if (t < 256) hist[t] = 0u;
    if (t < 8)   state[t] = (t == 1) ? (unsigned int)KSEL : 0u;
}

// ---------------------------------------------------------------------------
// alpha = questions @ question   via V_WMMA_F32_16X16X4_F32 (exact fp32)
//
// Per wave: one 16-row tile of `questions`, K swept 0..255 in steps of 4.
// A (16x4 f32, 2 VGPRs): lanes 0-15 row M=lane hold K=k0,k0+1;
//                        lanes 16-31 row M=lane-16 hold K=k0+2,k0+3.
// B (4x16 f32, 2 VGPRs): column-broadcast of question:
//                        VGPR0 = q[k0] (lanes 0-15) / q[k0+2] (lanes 16-31)
//                        VGPR1 = q[k0+1] / q[k0+3].
// D: VGPR r holds alpha[row0+r] in lanes 0-15 and alpha[row0+8+r] in 16-31;
//    all 16 columns identical, so lane 0 and lane 16 store 8 rows each.
// ---------------------------------------------------------------------------
__global__ void eernn_alpha_wmma(const float* __restrict__ questions,
                                 const float* __restrict__ question,
                                 float* __restrict__ alpha, int T) {
    __shared__ float qs[QDIM];
    int tid = threadIdx.x;
    for (int i = tid; i < QDIM; i += blockDim.x) qs[i] = question[i];
    __syncthreads();

    int lane    = tid & 31;
    int wave    = tid >> 5;
    int wpb     = blockDim.x >> 5;
    int gwave   = blockIdx.x * wpb + wave;
    int nwaves  = gridDim.x * wpb;
    int half    = lane >> 4;          // 0: lanes 0-15, 1: lanes 16-31
    int m       = lane & 15;
    int ntiles  = (T + 15) >> 4;

    for (int tile = gwave; tile < ntiles; tile += nwaves) {
        int row0 = tile << 4;
        int row  = row0 + m;
        if (row > T - 1) row = T - 1;          // clamp (T=100000 is 16-aligned)
        const float* arow = questions + (size_t)row * QDIM + (half << 1);

        v8f c = {};
        #pragma unroll 4
        for (int k0 = 0; k0 < QDIM; k0 += 4) {
            v2f a, b;
            a.x = arow[k0];
            a.y = arow[k0 + 1];
            b.x = qs[k0 + (half << 1)];
            b.y = qs[k0 + (half << 1) + 1];
            c = __builtin_amdgcn_wmma_f32_16x16x4_f32(
                    false, a, false, b, (short)0, c, false, false);
        }

        if ((lane & 15) == 0) {                // lane 0 and lane 16
            int base = row0 + (half << 3);
            #pragma unroll
            for (int r = 0; r < 8; ++r) {
                int rr = base + r;
                if (rr < T) alpha[rr] = c[r];
            }
        }
    }
}

// ---------------------------------------------------------------------------
// One radix-select histogram pass (MSB-first). Candidates = keys whose
// already-decided high bits (hiMask) match the prefix.
// ---------------------------------------------------------------------------
__global__ void eernn_hist(const float* __restrict__ alpha,
                           unsigned int* __restrict__ hist,
                           const unsigned int* __restrict__ state,
                           unsigned int hiMask, int shift, int T) {
    __shared__ unsigned int lh[256];
    int t = threadIdx.x;
    lh[t] = 0u;
    __syncthreads();

    unsigned int prefix = state[0] & hiMask;
    int i = blockIdx.x * blockDim.x + t;
    int stride = gridDim.x * blockDim.x;
    for (; i < T; i += stride) {
        unsigned int key = keyOf(alpha[i]);
        if ((key & hiMask) == prefix)
            atomicAdd(&lh[(key >> shift) & 0xFF], 1u);
    }
    __syncthreads();
    if (lh[t]) atomicAdd(&hist[t], lh[t]);
}

// ---------------------------------------------------------------------------
// Pick winning bin for this byte; update prefix / remainingK / countGreater;
// zero the histogram for the next pass.
// ---------------------------------------------------------------------------
__global__ void eernn_pick(unsigned int* __restrict__ hist,
                           unsigned int* __restrict__ state, int shift) {
    __shared__ unsigned int lh[256];
    int t = threadIdx.x;
    lh[t] = hist[t];
    __syncthreads();
    if (t == 0) {
        unsigned int remK = state[1];
        unsigned int cum = 0u;
        int chosen = 0;
        for (int b = 255; b >= 0; --b) {
            unsigned int c = lh[b];
            if (cum + c >= remK) { chosen = b; break; }
            cum += c;
        }
        state[0] |= ((unsigned int)chosen) << shift;
        state[1] = remK - cum;
        state[2] += cum;
    }
    __syncthreads();
    hist[t] = 0u;
}

// ---------------------------------------------------------------------------
// Compact the top-K: all keys > threshold, plus needEq keys == threshold.
// Slots [0,G) strictly-greater, [G, G+needEq) threshold ties. G+needEq == K.
// ---------------------------------------------------------------------------
__global__ void eernn_select(const float* __restrict__ alpha,
                             unsigned int* __restrict__ state,
                             float* __restrict__ selVals,
                             unsigned int* __restrict__ selIdx, int T) {
    unsigned int thresh = state[0];
    unsigned int needEq = state[1];
    unsigned int G      = state[2];
    int i = blockIdx.x * blockDim.x + threadIdx.x;
    int stride = gridDim.x * blockDim.x;
    for (; i < T; i += stride) {
        float v = alpha[i];
        unsigned int key = keyOf(v);
        if (key > thresh) {
            unsigned int p = atomicAdd(&state[3], 1u);
            selVals[p] = v;
            selIdx[p]  = (unsigned int)i;
        } else if (key == thresh) {
            unsigned int p = atomicAdd(&state[4], 1u);
            if (p < needEq) {
                selVals[G + p] = v;
                selIdx[G + p]  = (unsigned int)i;
            }
        }
    }
}

// ---------------------------------------------------------------------------
// Tail: softmax over selected vals, attention gather, score head, GRU step.
// Selections are re-sorted by index first so fp summation order is identical
// on every replay regardless of atomic compaction order.
// ---------------------------------------------------------------------------
__global__ void eernn_final(const float* __restrict__ question,
                            const float* __restrict__ score,
                            const float* __restrict__ hs,        // [T,1,64]
                            const float* __restrict__ W_score,   // [1,320]
                            const float* __restrict__ b_score,
                            const float* __restrict__ W_ih,      // [192,512]
                            const float* __restrict__ W_hh,      // [192,64]
                            const float* __restrict__ b_ih,
                            const float* __restrict__ b_hh,
                            const float* __restrict__ selVals,
                            const unsigned int* __restrict__ selIdx,
                            float* __restrict__ out, int T) {
    __shared__ float        vals[KSEL];
    __shared__ unsigned int idx[KSEL];
    __shared__ float        w[KSEL];
    __shared__ float        attn[HDIM];
    __shared__ float        red[256];
    __shared__ float        gi[3 * HDIM];
    __shared__ float        gh[3 * HDIM];
    __shared__ float        misc[2];

    int t = threadIdx.x;

    // deterministic order: sort selection by index (indices are unique)
    if (t < KSEL) {
        float v = selVals[t];
        unsigned int id = selIdx[t];
        int pos = 0;
        #pragma unroll 8
        for (int j = 0; j < KSEL; ++j) pos += (selIdx[j] < id) ? 1 : 0;
        vals[pos] = v;
        idx[pos]  = id;
    }
    __syncthreads();

    // softmax over the 64 selected scores
    if (t == 0) {
        float m = vals[0];
        for (int i = 1; i < KSEL; ++i) m = fmaxf(m, vals[i]);
        misc[0] = m;
    }
    __syncthreads();
    if (t < KSEL) w[t] = expf(vals[t] - misc[0]);
    __syncthreads();
    if (t == 0) {
        float s = 0.f;
        for (int i = 0; i < KSEL; ++i) s += w[i];
        misc[1] = s;
    }
    __syncthreads();

    // attn_h = softmax(vals) @ hs[idx]
    if (t < HDIM) {
        float inv = 1.0f / misc[1];
        float acc = 0.f;
        for (int i = 0; i < KSEL; ++i)
            acc += (w[i] * inv) * hs[(size_t)idx[i] * HDIM + t];
        attn[t] = acc;
    }
    __syncthreads();

    // pred = dot(W_score, [question ; attn_h]) + b_score
    {
        float p = W_score[t] * question[t];
        if (t < HDIM) p += W_score[QDIM + t] * attn[t];
        red[t] = p;
    }
    __syncthreads();
    for (int s = 128; s > 0; s >>= 1) {
        if (t < s) red[t] += red[t + s];
        __syncthreads();
    }
    if (t == 0) out[0] = red[0] + b_score[0];

    // gated GRU step: x = [q*gate ; q*(1-gate)] -> pick active half of W_ih
    const float* hlast = hs + (size_t)(T - 1) * HDIM;
    int off = (score[0] >= 0.5f) ? 0 : QDIM;
    if (t < 3 * HDIM) {
        float a = b_ih[t];
        const float* wr = W_ih + (size_t)t * (2 * QDIM) + off;
        for (int k = 0; k < QDIM; ++k) a += wr[k] * question[k];
        gi[t] = a;
        float b = b_hh[t];
        const float* wh = W_hh + (size_t)t * HDIM;
        for (int k = 0; k < HDIM; ++k) b += wh[k] * hlast[k];
        gh[t] = b;
    }
    __syncthreads();
    if (t < HDIM) {
        float r = 1.f / (1.f + expf(-(gi[t] + gh[t])));
        float z = 1.f / (1.f + expf(-(gi[HDIM + t] + gh[HDIM + t])));
        float n = tanhf(gi[2 * HDIM + t] + r * gh[2 * HDIM + t]);
        out[1 + t] = (1.f - z) * n + z * hlast[t];
    }
}

// ---------------------------------------------------------------------------
// Launcher. Workspace layout (needs ~402 KB):
//   [0, T*4)            alpha (f32)
//   +aligned: 1024 B    hist (256 x u32)
//   +32 B               state (8 x u32)
//   +256 B              selVals (64 x f32)
//   +256 B              selIdx  (64 x u32)
// ---------------------------------------------------------------------------
extern "C" void kernel_launch(void* const* d_in, const int* in_sizes, int n_in,
                              void* d_out, int out_size, void* d_ws, size_t ws_size,
                              hipStream_t stream) {
    const float* question  = (const float*)d_in[0];
    const float* score     = (const float*)d_in[1];
    const float* questions = (const float*)d_in[2];
    const float* hs        = (const float*)d_in[3];
    const float* W_score   = (const float*)d_in[4];
    const float* b_score   = (const float*)d_in[5];
    const float* W_ih      = (const float*)d_in[6];
    const float* W_hh      = (const float*)d_in[7];
    const float* b_ih      = (const float*)d_in[8];
    const float* b_hh      = (const float*)d_in[9];

    const int T = in_sizes[2] / QDIM;    // 100000

    char* ws = (char*)d_ws;
    size_t off = (size_t)T * sizeof(float);
    off = (off + 255) & ~(size_t)255;
    float*        alpha   = (float*)ws;
    unsigned int* hist    = (unsigned int*)(ws + off);           off += 1024;
    unsigned int* state   = (unsigned int*)(ws + off);           off += 32;
    float*        selVals = (float*)(ws + off);                  off += 256;
    unsigned int* selIdx  = (unsigned int*)(ws + off);

    float* out = (float*)d_out;          // out[0]=pred, out[1..64]=h_new

    // 1) init select state
    eernn_init<<<1, 256, 0, stream>>>(hist, state);

    // 2) bandwidth-bound alpha mat-vec on f32 WMMA
    eernn_alpha_wmma<<<256, 256, 0, stream>>>(questions, question, alpha, T);

    // 3) 4-pass MSB-first radix select of the top-64 threshold
    const unsigned int hiMasks[4] = {0x00000000u, 0xFF000000u, 0xFFFF0000u, 0xFFFFFF00u};
    const int          shifts [4] = {24, 16, 8, 0};
    for (int p = 0; p < 4; ++p) {
        eernn_hist<<<64, 256, 0, stream>>>(alpha, hist, state, hiMasks[p], shifts[p], T);
        eernn_pick<<<1, 256, 0, stream>>>(hist, state, shifts[p]);
    }

    // 4) compact the exact top-64 (values + indices)
    eernn_select<<<64, 256, 0, stream>>>(alpha, state, selVals, selIdx, T);

    // 5) softmax + attention + score head + GRU tail
    eernn_final<<<1, 256, 0, stream>>>(question, score, hs, W_score, b_score,
                                       W_ih, W_hh, b_ih, b_hh,
                                       selVals, selIdx, out, T);
}